// PeakNodeEmbedding_69063074120419
// MI455X (gfx1250) — compile-verified
//
#include <hip/hip_runtime.h>

#define BS    8
#define CCH   128
#define VV    5
#define HH    224
#define WW    224
#define KPTS  2048
#define DD    64
#define VHW   (VV*HH*WW)        // 250880
#define TM    64                // sample rows per block
#define SA_ST 132               // padded LDS stride for A (132 % 64 == 4)
#define SH_ST 68                // padded LDS stride for h / out tile

typedef float v2f __attribute__((ext_vector_type(2)));
typedef float v8f __attribute__((ext_vector_type(8)));

__global__ __launch_bounds__(128) void PeakNodeEmbedding_kernel(
    const float* __restrict__ fm,    // [BS,C,V,H,W]
    const float* __restrict__ peaks, // [BS,K,3]
    const float* __restrict__ W1,    // [C,D]
    const float* __restrict__ B1,    // [D]
    const float* __restrict__ W2,    // [D,D]
    const float* __restrict__ B2,    // [D]
    float* __restrict__ out)         // [BS,K,D]
{
    __shared__ float sA[TM * SA_ST];   // gathered features; reused as out tile
    __shared__ float sH[TM * SH_ST];   // hidden layer tile
    __shared__ int   sBase[TM];
    __shared__ int   sDx[TM];
    __shared__ int   sDyW[TM];
    __shared__ float sWx[TM], sWy[TM];
    __shared__ float sNrm[TM];

    const int tid = threadIdx.x;
    const int s0  = blockIdx.x * TM;

    // ---------------- per-sample bilinear parameters ----------------
    if (tid < TM) {
        const int g = s0 + tid;
        const int b = g / KPTS;
        const float pv = peaks[g*3 + 0];
        const float py = peaks[g*3 + 1];
        const float px = peaks[g*3 + 2];
        int v = (int)pv; v = v < 0 ? 0 : (v > VV-1 ? VV-1 : v);
        const float y0f = floorf(py), x0f = floorf(px);
        int y0 = (int)y0f; y0 = y0 < 0 ? 0 : (y0 > HH-1 ? HH-1 : y0);
        int x0 = (int)x0f; x0 = x0 < 0 ? 0 : (x0 > WW-1 ? WW-1 : x0);
        const int y1 = (y0 + 1 > HH-1) ? HH-1 : y0 + 1;
        const int x1 = (x0 + 1 > WW-1) ? WW-1 : x0 + 1;
        sBase[tid] = b * (CCH*VHW) + (v*HH + y0)*WW + x0;
        sDx[tid]   = x1 - x0;
        sDyW[tid]  = (y1 - y0) * WW;
        sWx[tid]   = px - x0f;
        sWy[tid]   = py - y0f;
    }
    __syncthreads();

    // ---------------- bilinear gather: thread == channel ----------------
    {
        const long coff = (long)tid * VHW;   // blockDim.x == CCH
        for (int s = 0; s < TM; ++s) {
            const float* p = fm + (long)sBase[s] + coff;
            const int dx = sDx[s], dy = sDyW[s];
            const float f00 = p[0], f01 = p[dx], f10 = p[dy], f11 = p[dy + dx];
            const float wx = sWx[s], wy = sWy[s];
            const float top = f00 + wx * (f01 - f00);
            const float bot = f10 + wx * (f11 - f10);
            sA[s * SA_ST + tid] = top + wy * (bot - top);
        }
    }
    __syncthreads();

    // ---------------- WMMA fragment indexing ----------------
    const int wave = tid >> 5;
    const int lane = tid & 31;
    const int m    = lane & 15;      // N index in B/C/D frags, M index in A frag
    const int half = lane >> 4;      // K-half for A/B frags, M-half for C/D frags
    const int rowA = 16 * wave + m;  // A-matrix row this lane supplies

    // ---------------- GEMM1: h = relu(A[64x128] * W1[128x64] + b1) ----------------
    v8f a0 = {}, a1 = {}, a2 = {}, a3 = {};
    for (int kk = 0; kk < CCH / 4; ++kk) {
        const int k0 = kk * 4 + 2 * half;
        v2f af;
        af.x = sA[rowA * SA_ST + k0];
        af.y = sA[rowA * SA_ST + k0 + 1];
        const float* bp = W1 + k0 * DD + m;
        v2f b0, b1v, b2v, b3v;
        b0.x  = bp[0];   b0.y  = bp[DD];
        b1v.x = bp[16];  b1v.y = bp[DD + 16];
        b2v.x = bp[32];  b2v.y = bp[DD + 32];
        b3v.x = bp[48];  b3v.y = bp[DD + 48];
        a0 = __builtin_amdgcn_wmma_f32_16x16x4_f32(false, af, false, b0,  (short)0, a0, false, false);
        a1 = __builtin_amdgcn_wmma_f32_16x16x4_f32(false, af, false, b1v, (short)0, a1, false, false);
        a2 = __builtin_amdgcn_wmma_f32_16x16x4_f32(false, af, false, b2v, (short)0, a2, false, false);
        a3 = __builtin_amdgcn_wmma_f32_16x16x4_f32(false, af, false, b3v, (short)0, a3, false, false);
    }
    {
        const float c0 = B1[m], c1 = B1[16 + m], c2 = B1[32 + m], c3 = B1[48 + m];
        for (int r = 0; r < 8; ++r) {
            const int row = 16 * wave + r + 8 * half;   // D-fragment M index
            float* hp = &sH[row * SH_ST + m];
            hp[0]  = fmaxf(a0[r] + c0, 0.0f);
            hp[16] = fmaxf(a1[r] + c1, 0.0f);
            hp[32] = fmaxf(a2[r] + c2, 0.0f);
            hp[48] = fmaxf(a3[r] + c3, 0.0f);
        }
    }
    __syncthreads();

    // ---------------- GEMM2: o = h[64x64] * W2[64x64] + b2 ----------------
    v8f o0 = {}, o1 = {}, o2 = {}, o3 = {};
    for (int kk = 0; kk < DD / 4; ++kk) {
        const int k0 = kk * 4 + 2 * half;
        v2f af;
        af.x = sH[rowA * SH_ST + k0];
        af.y = sH[rowA * SH_ST + k0 + 1];
        const float* bp = W2 + k0 * DD + m;
        v2f b0, b1v, b2v, b3v;
        b0.x  = bp[0];   b0.y  = bp[DD];
        b1v.x = bp[16];  b1v.y = bp[DD + 16];
        b2v.x = bp[32];  b2v.y = bp[DD + 32];
        b3v.x = bp[48];  b3v.y = bp[DD + 48];
        o0 = __builtin_amdgcn_wmma_f32_16x16x4_f32(false, af, false, b0,  (short)0, o0, false, false);
        o1 = __builtin_amdgcn_wmma_f32_16x16x4_f32(false, af, false, b1v, (short)0, o1, false, false);
        o2 = __builtin_amdgcn_wmma_f32_16x16x4_f32(false, af, false, b2v, (short)0, o2, false, false);
        o3 = __builtin_amdgcn_wmma_f32_16x16x4_f32(false, af, false, b3v, (short)0, o3, false, false);
    }
    {
        const float c0 = B2[m], c1 = B2[16 + m], c2 = B2[32 + m], c3 = B2[48 + m];
        for (int r = 0; r < 8; ++r) {
            const int row = 16 * wave + r + 8 * half;
            float* op = &sA[row * SH_ST + m];   // reuse sA as out tile (64x68 fits)
            op[0]  = o0[r] + c0;
            op[16] = o1[r] + c1;
            op[32] = o2[r] + c2;
            op[48] = o3[r] + c3;
        }
    }
    __syncthreads();

    // ---------------- per-row L2 normalization ----------------
    if (tid < TM) {
        const float* rp = &sA[tid * SH_ST];
        float ss = 0.0f;
        for (int j = 0; j < DD; ++j) { const float x = rp[j]; ss += x * x; }
        sNrm[tid] = 1.0f / fmaxf(sqrtf(ss), 1e-12f);
    }
    __syncthreads();

    // ---------------- coalesced writeback ----------------
    for (int i = tid; i < TM * DD; i += 128) {
        const int row = i >> 6;
        const int col = i & 63;
        out[(size_t)(s0 + row) * DD + col] = sA[row * SH_ST + col] * sNrm[row];
    }
}

extern "C" void kernel_launch(void* const* d_in, const int* in_sizes, int n_in,
                              void* d_out, int out_size, void* d_ws, size_t ws_size,
                              hipStream_t stream) {
    const float* fm    = (const float*)d_in[0];
    const float* peaks = (const float*)d_in[1];
    const float* W1    = (const float*)d_in[2];
    const float* B1    = (const float*)d_in[3];
    const float* W2    = (const float*)d_in[4];
    const float* B2    = (const float*)d_in[5];
    float* out = (float*)d_out;

    const int nblocks = (BS * KPTS) / TM;   // 256
    PeakNodeEmbedding_kernel<<<nblocks, 128, 0, stream>>>(fm, peaks, W1, B1, W2, B2, out);
}